// Voxelizer_50792283242970
// MI455X (gfx1250) — compile-verified
//
#include <hip/hip_runtime.h>
#include <hip/hip_bf16.h>

typedef __attribute__((ext_vector_type(2))) float v2f;
typedef __attribute__((ext_vector_type(8))) float v8f;

#define VD 128
#define VH 128
#define VW 128

__global__ void Voxelizer_zero_kernel(float* __restrict__ out, int n) {
    int i = blockIdx.x * blockDim.x + threadIdx.x;
    if (i < n) out[i] = 0.0f;
}

// One wave32 per tile of 16 Gaussians.
//   mahal(g, off) = coeff[g][0..9] . basis[off][0..9]   (K padded to 12)
// computed as 3 chained V_WMMA_F32_16X16X4_F32 per 16-offset tile (22 tiles cover 343 offsets).
__global__ __launch_bounds__(256) void Voxelizer_splat_kernel(
    const float* __restrict__ positions,   // (N,3)
    const float* __restrict__ scales,      // (N,3)
    const float* __restrict__ rotations,   // (N,4)
    const float* __restrict__ density,     // (N,)
    float* __restrict__ vol,               // (128^3,)
    int N, int ntiles) {

    const int lane = threadIdx.x & 31;
    const int wave = threadIdx.x >> 5;
    const int tile = blockIdx.x * (blockDim.x >> 5) + wave;
    if (tile >= ntiles) return;            // wave-uniform: EXEC stays all-ones for WMMA

    const int m  = lane & 15;              // gaussian-within-tile this lane's A-rows use / offset column
    const int hi = lane >> 4;              // lane half selects K-pair {0,1} vs {2,3} in f32 WMMA frags

    int g  = tile * 16 + m;
    int gi = g < N ? g : N - 1;

    // ---- per-gaussian coefficient computation (all lanes; lanes 16-31 mirror 0-15) ----
    float p0 = positions[3*gi+0], p1 = positions[3*gi+1], p2 = positions[3*gi+2];
    float s0 = scales[3*gi+0],    s1 = scales[3*gi+1],    s2 = scales[3*gi+2];
    float q0 = rotations[4*gi+0], q1 = rotations[4*gi+1];
    float q2 = rotations[4*gi+2], q3 = rotations[4*gi+3];
    float dens = density[gi];

    // voxel-space center (shape 128 -> half-extent 64)
    float cz = (p0 + 1.0f) * 64.0f - 0.5f;
    float cy = (p1 + 1.0f) * 64.0f - 0.5f;
    float cx = (p2 + 1.0f) * 64.0f - 0.5f;

    // quaternion -> rotation (matches reference normalization)
    float nrm = sqrtf(q0*q0 + q1*q1 + q2*q2 + q3*q3) + 1e-12f;
    float w = q0/nrm, x = q1/nrm, y = q2/nrm, z = q3/nrm;
    float R00 = 1.0f-2.0f*(y*y+z*z), R01 = 2.0f*(x*y-w*z),     R02 = 2.0f*(x*z+w*y);
    float R10 = 2.0f*(x*y+w*z),      R11 = 1.0f-2.0f*(x*x+z*z), R12 = 2.0f*(y*z-w*x);
    float R20 = 2.0f*(x*z-w*y),      R21 = 2.0f*(y*z+w*x),      R22 = 1.0f-2.0f*(x*x+y*y);

    float i0 = 1.0f/(s0+1e-8f), i1 = 1.0f/(s1+1e-8f), i2 = 1.0f/(s2+1e-8f);
    float L00 = R00*i0, L01 = R01*i1, L02 = R02*i2;
    float L10 = R10*i0, L11 = R11*i1, L12 = R12*i2;
    float L20 = R20*i0, L21 = R21*i1, L22 = R22*i2;

    // A = (L L^T) / 64^2  (diff is divided by half-shape = 64 in the reference)
    const float kk = 1.0f / 4096.0f;
    float A00 = (L00*L00 + L01*L01 + L02*L02) * kk;
    float A11 = (L10*L10 + L11*L11 + L12*L12) * kk;
    float A22 = (L20*L20 + L21*L21 + L22*L22) * kk;
    float A01 = (L00*L10 + L01*L11 + L02*L12) * kk;
    float A02 = (L00*L20 + L01*L21 + L02*L22) * kk;
    float A12 = (L10*L20 + L11*L21 + L12*L22) * kk;

    float rz = rintf(cz), ry = rintf(cy), rx = rintf(cx);   // RNE == jnp.round
    float fz = cz - rz,   fy = cy - ry,   fx = cx - rx;
    int bz = (int)rz, by = (int)ry, bx = (int)rx;

    float lz = -2.0f*(A00*fz + A01*fy + A02*fx);
    float ly = -2.0f*(A01*fz + A11*fy + A12*fx);
    float lx = -2.0f*(A02*fz + A12*fy + A22*fx);
    float c0 = -0.5f*(lz*fz + ly*fy + lx*fx);               // = f^T A f

    if (g >= N) { dens = 0.0f; bz = -1000000; by = -1000000; bx = -1000000; }

    // coeff order paired with basis {oz^2,oy^2,ox^2, oz*oy,oz*ox,oy*ox, oz,oy,ox, 1, 0,0}
    float k0 = A00,        k1 = A11,        k2 = A22;
    float k3 = 2.0f*A01,   k4 = 2.0f*A02,   k5 = 2.0f*A12;
    float k6 = lz,         k7 = ly,         k8 = lx;
    float k9 = c0,         k10 = 0.0f,      k11 = 0.0f;

    // ---- A fragments (documented f32 16x4 layout: VGPR0=K0|K2, VGPR1=K1|K3 per lane half) ----
    v2f a0, a1, a2;
    a0.x = hi ? k2  : k0;   a0.y = hi ? k3  : k1;
    a1.x = hi ? k6  : k4;   a1.y = hi ? k7  : k5;
    a2.x = hi ? k10 : k8;   a2.y = hi ? k11 : k9;

    // ---- metadata for the 8 D-rows this lane will hold (gaussians v + 8*hi) ----
    float dv[8]; int bzv[8], byv[8], bxv[8];
#pragma unroll
    for (int v = 0; v < 8; ++v) {
        int src = v + (hi << 3);
        dv[v]  = __shfl(dens, src, 32);
        bzv[v] = __shfl(bz,   src, 32);
        byv[v] = __shfl(by,   src, 32);
        bxv[v] = __shfl(bx,   src, 32);
    }

    // ---- 22 offset tiles of 16 (343 offsets padded to 352) ----
    for (int jt = 0; jt < 22; ++jt) {
        int j   = jt * 16 + m;                 // this lane's offset column
        int izq = j / 49;
        int rem = j - izq * 49;
        int iyq = rem / 7;
        int ixq = rem - iyq * 7;
        int oz = izq - 3, oy = iyq - 3, ox = ixq - 3;
        float ozf = (float)oz, oyf = (float)oy, oxf = (float)ox;

        float b0v = ozf*ozf, b1v = oyf*oyf, b2v = oxf*oxf;
        float b3v = ozf*oyf, b4v = ozf*oxf, b5v = oyf*oxf;
        float b6v = ozf,     b7v = oyf,     b8v = oxf;
        float b9v = 1.0f,    b10v = 0.0f,   b11v = 0.0f;

        // B fragments: 4x16 f32, transpose-symmetric to the A layout
        v2f b0, b1, b2;
        b0.x = hi ? b2v  : b0v;   b0.y = hi ? b3v  : b1v;
        b1.x = hi ? b6v  : b4v;   b1.y = hi ? b7v  : b5v;
        b2.x = hi ? b10v : b8v;   b2.y = hi ? b11v : b9v;

        v8f c = {0.f,0.f,0.f,0.f,0.f,0.f,0.f,0.f};
        c = __builtin_amdgcn_wmma_f32_16x16x4_f32(false, a0, false, b0, (short)0, c, false, false);
        c = __builtin_amdgcn_wmma_f32_16x16x4_f32(false, a1, false, b1, (short)0, c, false, false);
        c = __builtin_amdgcn_wmma_f32_16x16x4_f32(false, a2, false, b2, (short)0, c, false, false);

        bool jvalid = (j < 343);
#pragma unroll
        for (int v = 0; v < 8; ++v) {
            float mah = c[v];                  // gaussian v + 8*hi, offset column m
            int gz = bzv[v] + oz;
            int gy = byv[v] + oy;
            int gx = bxv[v] + ox;
            bool ok = jvalid && (mah <= 9.0f) &&
                      (gz >= 0) && (gz < VD) &&
                      (gy >= 0) && (gy < VH) &&
                      (gx >= 0) && (gx < VW);
            if (ok) {
                float val = __expf(-0.5f * mah) * dv[v];
                float* addr = vol + ((gz * VH + gy) * VW + gx);
                (void)__hip_atomic_fetch_add(addr, val, __ATOMIC_RELAXED,
                                             __HIP_MEMORY_SCOPE_AGENT);
            }
        }
    }
}

extern "C" void kernel_launch(void* const* d_in, const int* in_sizes, int n_in,
                              void* d_out, int out_size, void* d_ws, size_t ws_size,
                              hipStream_t stream) {
    const float* positions = (const float*)d_in[0];
    const float* scales    = (const float*)d_in[1];
    const float* rotations = (const float*)d_in[2];
    const float* density   = (const float*)d_in[3];
    float* vol = (float*)d_out;
    int N = in_sizes[3];

    int nvox = VD * VH * VW;
    Voxelizer_zero_kernel<<<(nvox + 255) / 256, 256, 0, stream>>>(vol, nvox);

    int ntiles = (N + 15) / 16;
    const int waves_per_block = 8;                    // 256 threads = 8 wave32
    int blocks = (ntiles + waves_per_block - 1) / waves_per_block;
    Voxelizer_splat_kernel<<<blocks, waves_per_block * 32, 0, stream>>>(
        positions, scales, rotations, density, vol, N, ntiles);
}